// Transducer_33689723470022
// MI455X (gfx1250) — compile-verified
//
#include <hip/hip_runtime.h>
#include <hip/hip_bf16.h>

// Problem constants (fixed by the reference's setup_inputs()).
constexpr int kB = 8;
constexpr int kT = 2048;
constexpr int kC = 4096;
constexpr int kU = 200;
constexpr float kNEG = -1e30f;

// ---------------------------------------------------------------------------
// Kernel 1: fused log-softmax statistics + target gather.
//   One workgroup (256 thr = 8 wave32) per (b,t) row of C=4096 floats.
//   Row is staged global->LDS with CDNA5 async loads
//   (global_load_async_to_lds_b128, tracked by ASYNCcnt), because the row is
//   consumed twice: streaming reduction (max, sum-exp) AND random gather at
//   200 target columns. 256 MB total traffic -> ~11 us at 23.3 TB/s; the
//   ~67M v_exp_f32 hide well under that across ~1k SIMD32s.
// ---------------------------------------------------------------------------
__global__ __launch_bounds__(256) void tx_softmax_gather(
    const float* __restrict__ x,        // [B, T, C]
    const int* __restrict__ tgt,        // [B, U]
    float* __restrict__ lpt) {          // [B, T, U] workspace
  __shared__ float row[kC];             // 16 KB of the 320 KB/WGP pool
  __shared__ float redmax[8];
  __shared__ float redsum[8];

  const int tid = threadIdx.x;
  const int r   = blockIdx.x;           // flat b*T + t
  const int b   = r / kT;

  // --- async DMA: global row -> LDS, 4 chunks, 16B per thread per chunk ---
  const float* xrow = x + (size_t)r * kC;                // uniform (SGPR) base
  const uint32_t lds0 = (uint32_t)(uintptr_t)(&row[0]);  // low 32 bits = LDS offset
#pragma unroll
  for (int k = 0; k < 4; ++k) {
    uint32_t voff = (uint32_t)(tid * 16 + k * 4096);     // byte offset in row
    uint32_t ldsa = lds0 + voff;
    asm volatile("global_load_async_to_lds_b128 %0, %1, %2"
                 :: "v"(ldsa), "v"(voff), "s"(xrow)
                 : "memory");
  }
  asm volatile("s_wait_asynccnt 0x0" ::: "memory");
  __syncthreads();

  // --- pass 1: row max (bank-conflict-free strided reads) ---
  float lmax = -3.4e38f;
#pragma unroll
  for (int k = 0; k < 16; ++k) lmax = fmaxf(lmax, row[tid + 256 * k]);
#pragma unroll
  for (int off = 16; off > 0; off >>= 1)
    lmax = fmaxf(lmax, __shfl_xor(lmax, off, 32));
  if ((tid & 31) == 0) redmax[tid >> 5] = lmax;
  __syncthreads();
  float rmax = redmax[0];
#pragma unroll
  for (int w = 1; w < 8; ++w) rmax = fmaxf(rmax, redmax[w]);

  // --- pass 2: sum exp(x - max) ---
  float lsum = 0.f;
#pragma unroll
  for (int k = 0; k < 16; ++k) lsum += __expf(row[tid + 256 * k] - rmax);
#pragma unroll
  for (int off = 16; off > 0; off >>= 1) lsum += __shfl_xor(lsum, off, 32);
  if ((tid & 31) == 0) redsum[tid >> 5] = lsum;
  __syncthreads();
  float rsum = 0.f;
#pragma unroll
  for (int w = 0; w < 8; ++w) rsum += redsum[w];
  const float lse = rmax + __logf(rsum);

  // --- gather per-target log-probs straight out of LDS ---
  if (tid < kU) {
    const int c = tgt[b * kU + tid];
    lpt[(size_t)r * kU + tid] = row[c] - lse;
  }
}

// stable logaddexp, consistent with jnp.logaddexp under the -1e30 sentinel
__device__ __forceinline__ float log_add_exp(float p, float q) {
  float m = fmaxf(p, q);
  float d = fabsf(p - q);              // huge/inf -> expf gives 0 -> returns m
  return m + __logf(1.0f + __expf(-d));
}

// ---------------------------------------------------------------------------
// Kernel 2: alpha lattice scan. Serial in T (2047-step dependency chain),
// parallel only across U=200. One wave32 per batch, 7 u-values per lane
// (u = lane + 32k), so there are ZERO barriers on the critical path: the
// alpha[u-1] neighbor is a __shfl_up plus a lane-31 wrap broadcast. The 7
// logaddexps per step are independent and pipeline through the trans units.
// The lpt row loads are recurrence-independent, so they are software-
// pipelined one row ahead (load t+1 overlaps compute t) and the t+2 row is
// prefetched with full 1 KB coverage (lane*32B stride spans the 800B row).
// ---------------------------------------------------------------------------
__global__ __launch_bounds__(32) void tx_alpha_scan(
    const float* __restrict__ lpt,      // [B, T, U]
    float* __restrict__ part) {         // [B] forward scores
  const int b    = blockIdx.x;
  const int lane = threadIdx.x;
  const float* lp = lpt + (size_t)b * kT * kU;

  float a[7];                           // alpha at step t-1
#pragma unroll
  for (int k = 0; k < 7; ++k) {
    const int u = lane + 32 * k;
    a[k] = (u == 0) ? lp[0] : kNEG;     // alpha0
  }

  float l[7];                           // lpt row for the current step
  {
    const float* lrow = lp + (size_t)kU;        // row t=1
#pragma unroll
    for (int k = 0; k < 7; ++k) {
      const int u = lane + 32 * k;
      l[k] = (u < kU) ? lrow[u] : kNEG;
    }
  }

  for (int t = 1; t < kT; ++t) {
    // issue next row's loads + prefetch BEFORE the dependent trans chain
    float ln[7];
#pragma unroll
    for (int k = 0; k < 7; ++k) ln[k] = kNEG;
    if (t + 1 < kT) {
      const float* nrow = lp + (size_t)(t + 1) * kU;
      // prefetch row t+2: 32 lanes x 32B stride = 1024B >= 800B row
      __builtin_prefetch((const char*)(nrow + kU) + lane * 32, 0, 0);
#pragma unroll
      for (int k = 0; k < 7; ++k) {
        const int u = lane + 32 * k;
        if (u < kU) ln[k] = nrow[u];
      }
    }

    float na[7];
#pragma unroll
    for (int k = 0; k < 7; ++k) {
      const float up   = __shfl_up(a[k], 1, 32);             // lane i-1 -> i
      const float wrap = __shfl(a[k == 0 ? 0 : k - 1], 31, 32);
      const float prev = (lane == 0) ? ((k == 0) ? kNEG : wrap) : up;
      na[k] = l[k] + log_add_exp(a[k], prev);
    }
#pragma unroll
    for (int k = 0; k < 7; ++k) { a[k] = na[k]; l[k] = ln[k]; }
  }

  // u = U-1 = 199 = lane 7 + 32*6
  if (lane == 7) part[b] = a[6];
}

// ---------------------------------------------------------------------------
// Kernel 3: loss = mean_b(-alpha[T-1, U-1]). Deterministic, no atomics into
// the (poisoned, never re-initialized) d_out.
// ---------------------------------------------------------------------------
__global__ void tx_finalize(const float* __restrict__ part,
                            float* __restrict__ out) {
  if (threadIdx.x == 0 && blockIdx.x == 0) {
    float s = 0.f;
    for (int b = 0; b < kB; ++b) s += part[b];
    out[0] = -s / (float)kB;
  }
}

extern "C" void kernel_launch(void* const* d_in, const int* in_sizes, int n_in,
                              void* d_out, int out_size, void* d_ws, size_t ws_size,
                              hipStream_t stream) {
  (void)in_sizes; (void)n_in; (void)out_size; (void)ws_size;
  const float* x   = (const float*)d_in[0];   // [B,T,C] float32
  const int*   tgt = (const int*)d_in[1];     // [B,U] integer targets
  float*       out = (float*)d_out;           // scalar loss

  // workspace: lpt [B,T,U] floats, then 8 per-batch partials
  float* lpt  = (float*)d_ws;
  float* part = (float*)((char*)d_ws + (size_t)kB * kT * kU * sizeof(float));

  tx_softmax_gather<<<kB * kT, 256, 0, stream>>>(x, tgt, lpt);
  tx_alpha_scan<<<kB, 32, 0, stream>>>(lpt, part);
  tx_finalize<<<1, 32, 0, stream>>>(part, out);
}